// LSTMWithMetadata_7945689497854
// MI455X (gfx1250) — compile-verified
//
#include <hip/hip_runtime.h>
#include <cstddef>
#include <cstdint>

// ---------------------------------------------------------------------------
// MI455X (gfx1250) implementation of the BiLSTM + FC reference.
//  - All GEMMs use V_WMMA_F32_16X16X4_F32 (fp32 WMMA; K is tiny, accuracy
//    matters through 1024 recurrent steps, and the chip is bandwidth-bound
//    here anyway: ~2.4GB of pre-activation traffic @ 23.3TB/s ~= 100us).
//  - FC GEMM weight tiles are staged into LDS by the Tensor Data Mover
//    (tensor_load_to_lds + s_wait_tensorcnt), with TDM pad_interval/amount
//    reproducing the +1-DWORD bank-conflict padding and TDM OOB-row zeroing
//    handling the N=40 tail of fc4.
//  - Recurrence parallelized over batch (independent per b): 1 thread per
//    batch element, t-loop in registers, Whh broadcast from LDS,
//    global_prefetch_b8 on next timestep's gate row.
// ---------------------------------------------------------------------------

typedef float v2f __attribute__((ext_vector_type(2)));
typedef float v8f __attribute__((ext_vector_type(8)));
typedef unsigned int u32x4 __attribute__((ext_vector_type(4)));
typedef int i32x4 __attribute__((ext_vector_type(4)));
typedef int i32x8 __attribute__((ext_vector_type(8)));

#define TT 1024   // sequence length
#define BB 1024   // batch
#define HH 10     // hidden
#define GG 40     // 4*H gates
#define EPS 1e-5f

__device__ __forceinline__ v8f wmma_f32(v2f a, v2f b, v8f c) {
  // D = A(16x4 f32) * B(4x16 f32) + C(16x16 f32)
  return __builtin_amdgcn_wmma_f32_16x16x4_f32(
      /*neg_a=*/false, a, /*neg_b=*/false, b,
      /*c_mod=*/(short)0, c, /*reuse_a=*/false, /*reuse_b=*/false);
}

__device__ __forceinline__ float sigm(float x) {
  return 1.0f / (1.0f + __expf(-x));
}

// ---------------------------------------------------------------------------
// TDM: DMA a 2D tile (tile_dim0=16 dwords x tile_dim1=64 rows) of f32 from
// global (row stride = strideK elements) into LDS at lds_off, inserting 1
// padding DWORD after every 16 DWORDs (-> LDS rows of 17 floats).
// rem_k / rem_n are the remaining tensor extents measured from the tile
// origin; rows >= rem_n are out-of-bounds and written as zeros by the TDM.
// ---------------------------------------------------------------------------
__device__ __forceinline__ void tdm_load_w_tile(uint32_t lds_off,
                                                const float* gptr,
                                                int rem_k, int rem_n,
                                                int strideK) {
  const uint64_t ga = (uint64_t)(uintptr_t)gptr;
  u32x4 g0;
  g0[0] = 1u;                                    // count=1, user-mode D#
  g0[1] = lds_off;                               // lds_addr (bytes)
  g0[2] = (uint32_t)(ga & 0xFFFFFFFFu);          // global_addr[31:0]
  g0[3] = (uint32_t)((ga >> 32) & 0x01FFFFFFu)   // global_addr[56:32]
          | 0x80000000u;                         // type=2 ("image")
  i32x8 g1;
  g1[0] = (2 << 16)      // data_size = 4 bytes
        | (1 << 20)      // pad_enable
        | (3 << 22);     // pad_interval = 16 DWORDs (0=2,1=4,2=8,3=16)
                         // pad_amount  = 1 DWORD  (field value 0)
  g1[1] = (rem_k & 0xFFFF) << 16;                         // tensor_dim0 lo
  g1[2] = ((rem_k >> 16) & 0xFFFF) | ((rem_n & 0xFFFF) << 16); // dim0 hi | dim1 lo
  g1[3] = ((rem_n >> 16) & 0xFFFF) | (16 << 16);          // dim1 hi | tile_dim0=16
  g1[4] = 64;                                             // tile_dim1=64, tile_dim2=0
  g1[5] = strideK;                                        // tensor_dim0_stride lo
  g1[6] = 0;                                              // stride hi | dim1_stride lo
  g1[7] = 0;
  const i32x4 z4 = {0, 0, 0, 0};                          // groups 2/3 unused (2D tile)
  const i32x8 z8 = {0, 0, 0, 0, 0, 0, 0, 0};
  __builtin_amdgcn_tensor_load_to_lds(g0, g1, z4, z4, z8, 0);
}

// fused concat(xyz[B,T,3], shapes[B,T,9], amino[B,T,21]) -> x[t][b][k], k<33
__device__ __forceinline__ float load_concat(const float* __restrict__ xyz,
                                             const float* __restrict__ sh,
                                             const float* __restrict__ aa,
                                             int t, int b, int k) {
  if (k < 3)  return xyz[((size_t)b * TT + t) * 3 + k];
  if (k < 12) return sh [((size_t)b * TT + t) * 9 + (k - 3)];
  if (k < 33) return aa [((size_t)b * TT + t) * 21 + (k - 12)];
  return 0.0f;
}

// ---------------------------------------------------------------------------
// pre[t][b][g] = x[t][b][:] @ Wih[g,:]^T + bih[g] + bhh[g]
// One wave per (t, 16-batch) tile; 3 N-subtiles (40 gates) share the A frag.
// ---------------------------------------------------------------------------
__global__ __launch_bounds__(128) void pre_gemm_kernel(
    int mode, int K, int Kp,
    const float* __restrict__ xyz, const float* __restrict__ shapes,
    const float* __restrict__ amino, const float* __restrict__ Xin,
    const float* __restrict__ Wih, const float* __restrict__ bih,
    const float* __restrict__ bhh, float* __restrict__ pre) {
  __shared__ float sW[GG * 36];
  __shared__ float sB[GG];
  const int tid = threadIdx.x;
  for (int i = tid; i < GG * Kp; i += 128) {
    int g = i / Kp, k = i % Kp;
    sW[i] = (k < K) ? Wih[g * K + k] : 0.0f;
  }
  for (int i = tid; i < GG; i += 128) sB[i] = bih[i] + bhh[i];
  __syncthreads();

  const int wave = tid >> 5, lane = tid & 31;
  const int half = lane >> 4, l16 = lane & 15;
  const int tile = blockIdx.x * 4 + wave;
  const int t = tile >> 6;
  const int b0 = (tile & 63) << 4;
  const int brow = b0 + l16;

  v8f acc0 = {0.f,0.f,0.f,0.f,0.f,0.f,0.f,0.f};
  v8f acc1 = acc0, acc2 = acc0;

  for (int k0 = 0; k0 < Kp; k0 += 4) {
    const int kb = k0 + 2 * half;
    v2f a;
    if (mode == 0) {
      a.x = load_concat(xyz, shapes, amino, t, brow, kb);
      a.y = load_concat(xyz, shapes, amino, t, brow, kb + 1);
    } else {
      const float* xr = Xin + ((size_t)t * BB + brow) * 20;
      a.x = (kb < K) ? xr[kb] : 0.0f;
      a.y = (kb + 1 < K) ? xr[kb + 1] : 0.0f;
    }
    { v2f bf; bf.x = sW[l16 * Kp + kb];        bf.y = sW[l16 * Kp + kb + 1];
      acc0 = wmma_f32(a, bf, acc0); }
    { int g = 16 + l16;
      v2f bf; bf.x = sW[g * Kp + kb];          bf.y = sW[g * Kp + kb + 1];
      acc1 = wmma_f32(a, bf, acc1); }
    { int g = 32 + l16; int gc = (g < GG) ? g : 0;
      v2f bf; bf.x = sW[gc * Kp + kb];         bf.y = sW[gc * Kp + kb + 1];
      acc2 = wmma_f32(a, bf, acc2); }
  }

#pragma unroll
  for (int v = 0; v < 8; ++v) {
    const int m = 8 * half + v;
    const size_t row = ((size_t)t * BB + b0 + m) * GG;
    pre[row + l16]      = acc0[v] + sB[l16];
    pre[row + 16 + l16] = acc1[v] + sB[16 + l16];
    if (32 + l16 < GG) pre[row + 32 + l16] = acc2[v] + sB[32 + l16];
  }
}

// ---------------------------------------------------------------------------
// Recurrence: one thread per batch element. gates = pre[t,b,:] + h @ Whh^T.
// i,f,g,o torch gate order. Writes h into Xout[t][b][dir*10 + k].
// ---------------------------------------------------------------------------
__global__ __launch_bounds__(256) void recur_kernel(
    const float* __restrict__ pre, const float* __restrict__ Whh,
    float* __restrict__ Xout, int dir) {
  __shared__ float sW[GG * HH];
  for (int i = threadIdx.x; i < GG * HH; i += 256) sW[i] = Whh[i];
  __syncthreads();

  const int b = blockIdx.x * 256 + threadIdx.x;
  float h[HH], c[HH];
#pragma unroll
  for (int k = 0; k < HH; ++k) { h[k] = 0.0f; c[k] = 0.0f; }

  for (int s = 0; s < TT; ++s) {
    const int t = dir ? (TT - 1 - s) : s;
    const float* pr = pre + ((size_t)t * BB + b) * GG;
    if (s + 1 < TT) {
      const int t2 = dir ? (TT - 2 - s) : (s + 1);
      __builtin_prefetch(pre + ((size_t)t2 * BB + b) * GG, 0, 1); // global_prefetch_b8
    }
    float g[GG];
#pragma unroll
    for (int q = 0; q < GG / 4; ++q) {
      float4 f4 = ((const float4*)pr)[q];
      g[4 * q] = f4.x; g[4 * q + 1] = f4.y; g[4 * q + 2] = f4.z; g[4 * q + 3] = f4.w;
    }
#pragma unroll
    for (int gi = 0; gi < GG; ++gi) {
      float acc = g[gi];
#pragma unroll
      for (int k = 0; k < HH; ++k) acc = fmaf(sW[gi * HH + k], h[k], acc);
      g[gi] = acc;
    }
#pragma unroll
    for (int k = 0; k < HH; ++k) {
      const float ig = sigm(g[k]);
      const float fg = sigm(g[HH + k]);
      const float gg = tanhf(g[2 * HH + k]);
      const float og = sigm(g[3 * HH + k]);
      c[k] = fg * c[k] + ig * gg;
      h[k] = og * tanhf(c[k]);
      Xout[((size_t)t * BB + b) * 20 + dir * HH + k] = h[k];
    }
  }
}

// ---------------------------------------------------------------------------
// proj[b][t*3+p] = x[t][b][:20] @ project[:,p]
// ---------------------------------------------------------------------------
__global__ __launch_bounds__(256) void proj_kernel(
    const float* __restrict__ X, const float* __restrict__ P,
    float* __restrict__ out) {
  const int idx = blockIdx.x * 256 + threadIdx.x;
  const int t = idx >> 10, b = idx & (BB - 1);
  const float* xr = X + ((size_t)t * BB + b) * 20;
  float xv[20];
#pragma unroll
  for (int q = 0; q < 5; ++q) {
    float4 f4 = ((const float4*)xr)[q];
    xv[4 * q] = f4.x; xv[4 * q + 1] = f4.y; xv[4 * q + 2] = f4.z; xv[4 * q + 3] = f4.w;
  }
#pragma unroll
  for (int p = 0; p < 3; ++p) {
    float acc = 0.0f;
#pragma unroll
    for (int k = 0; k < 20; ++k) acc = fmaf(xv[k], P[k * 3 + p], acc);
    out[(size_t)b * 3072 + t * 3 + p] = acc;
  }
}

// ---------------------------------------------------------------------------
// Generic WMMA-f32 FC: out[M,N] = epilogue( [A0|A1][M,K] @ W[N,K]^T )
// Block tile 32(M) x 64(N), 8 waves. W tiles are DMA'd into LDS by the TDM
// (wave 0 issues tensor_load_to_lds, waits TENSORcnt, barrier publishes);
// A tiles staged by all threads (carries the fc2 concat split).
// mode 0: ReLU -> BatchNorm(q0=g,q1=b,q2=m,q3=v);  mode 1: + bias(q0)
// ---------------------------------------------------------------------------
__global__ __launch_bounds__(256) void fc_gemm_kernel(
    const float* __restrict__ A0, int lda0, int cols0,
    const float* __restrict__ A1, int lda1,
    int K, const float* __restrict__ W, int N,
    float* __restrict__ out, int mode,
    const float* __restrict__ q0, const float* __restrict__ q1,
    const float* __restrict__ q2, const float* __restrict__ q3) {
  __shared__ float As[32][17];
  __shared__ float Ws[64][17];
  const int tid = threadIdx.x;
  const int wave = tid >> 5, lane = tid & 31;
  const int half = lane >> 4, l16 = lane & 15;
  const int wm = wave >> 2, wn = wave & 3;
  const int bm = blockIdx.x & 31, bn = blockIdx.x >> 5;
  const int m0 = bm * 32, n0 = bn * 64;
  const uint32_t ws_lds = (uint32_t)(uintptr_t)(&Ws[0][0]); // LDS byte offset
  const int rem_n = N - n0;                                  // rows beyond -> TDM zeros

  v8f acc = {0.f,0.f,0.f,0.f,0.f,0.f,0.f,0.f};

  for (int k0 = 0; k0 < K; k0 += 16) {
    // W tile 64x16 via Tensor Data Mover (wave 0 only; EXEC-independent)
    if (wave == 0) {
      tdm_load_w_tile(ws_lds, W + (size_t)n0 * K + k0, K - k0, rem_n, K);
    }
    // stage A tile 32x16
#pragma unroll
    for (int p = 0; p < 2; ++p) {
      const int idx = tid + p * 256;
      const int r = idx >> 4, cc = idx & 15;
      const int kk = k0 + cc;
      float v;
      if (kk < cols0) v = A0[(size_t)(m0 + r) * lda0 + kk];
      else            v = A1[(size_t)(m0 + r) * lda1 + (kk - cols0)];
      As[r][cc] = v;
    }
    if (wave == 0) __builtin_amdgcn_s_wait_tensorcnt(0);
    __syncthreads();
#pragma unroll
    for (int ks = 0; ks < 16; ks += 4) {
      const int kb = ks + 2 * half;
      v2f a, bf;
      a.x  = As[wm * 16 + l16][kb];  a.y  = As[wm * 16 + l16][kb + 1];
      bf.x = Ws[wn * 16 + l16][kb];  bf.y = Ws[wn * 16 + l16][kb + 1];
      acc = wmma_f32(a, bf, acc);
    }
    __syncthreads();
  }

#pragma unroll
  for (int v = 0; v < 8; ++v) {
    const int m = m0 + wm * 16 + 8 * half + v;
    const int n = n0 + wn * 16 + l16;
    if (n < N) {
      float x = acc[v];
      if (mode == 0) {
        x = fmaxf(x, 0.0f);
        x = q0[n] * (x - q2[n]) * rsqrtf(q3[n] + EPS) + q1[n];
      } else {
        x += q0[n];
      }
      out[(size_t)m * N + n] = x;
    }
  }
}

// ---------------------------------------------------------------------------
extern "C" void kernel_launch(void* const* d_in, const int* in_sizes, int n_in,
                              void* d_out, int out_size, void* d_ws, size_t ws_size,
                              hipStream_t stream) {
  (void)in_sizes; (void)n_in; (void)out_size; (void)ws_size;
  const float* xyz    = (const float*)d_in[0];
  const float* shapes = (const float*)d_in[1];
  const float* amino  = (const float*)d_in[2];
  const float* seqvec = (const float*)d_in[3];
  const float* Wih0   = (const float*)d_in[4];
  const float* Whh0   = (const float*)d_in[5];
  const float* bih0   = (const float*)d_in[6];
  const float* bhh0   = (const float*)d_in[7];
  const float* WihR   = (const float*)d_in[8];
  const float* WhhR   = (const float*)d_in[9];
  const float* bihR   = (const float*)d_in[10];
  const float* bhhR   = (const float*)d_in[11];
  const float* projW  = (const float*)d_in[12];
  const float* fc1_w  = (const float*)d_in[13];
  const float* bn1_g  = (const float*)d_in[14];
  const float* bn1_b  = (const float*)d_in[15];
  const float* bn1_m  = (const float*)d_in[16];
  const float* bn1_v  = (const float*)d_in[17];
  const float* fc2_w  = (const float*)d_in[18];
  const float* bn2_g  = (const float*)d_in[19];
  const float* bn2_b  = (const float*)d_in[20];
  const float* bn2_m  = (const float*)d_in[21];
  const float* bn2_v  = (const float*)d_in[22];
  const float* fc3_w  = (const float*)d_in[23];
  const float* bn3_g  = (const float*)d_in[24];
  const float* bn3_b  = (const float*)d_in[25];
  const float* bn3_m  = (const float*)d_in[26];
  const float* bn3_v  = (const float*)d_in[27];
  const float* fc4_w  = (const float*)d_in[28];
  const float* fc4_b  = (const float*)d_in[29];
  float* out = (float*)d_out;

  // workspace layout (floats): pre[T*B*40] | xA[T*B*20] | xB[T*B*20]
  // FC temporaries reuse the pre region after the LSTM is done.
  float* wsf   = (float*)d_ws;
  float* pre   = wsf;
  float* xA    = wsf + (size_t)TT * BB * GG;
  float* xB    = xA  + (size_t)TT * BB * 20;
  float* projb = pre;                      // [1024, 3072]
  float* y1    = pre + (size_t)BB * 3072;  // [1024, 256]
  float* lat   = y1  + (size_t)BB * 256;   // [1024, 1024]
  float* y3    = lat + (size_t)BB * 1024;  // [1024, 512]

  for (int l = 0; l < 3; ++l) {
    const float* Xin = (l == 0) ? nullptr : ((l == 1) ? xA : xB);
    float* Xout = (l == 1) ? xB : xA;       // l0->xA, l1->xB, l2->xA
    const int K = (l == 0) ? 33 : 20;
    const int Kp = (l == 0) ? 36 : 20;
    for (int dir = 0; dir < 2; ++dir) {
      const float *Wih, *Whh, *bih, *bhh;
      if (l == 0) {
        Wih = Wih0 + dir * GG * 33; Whh = Whh0 + dir * GG * HH;
        bih = bih0 + dir * GG;      bhh = bhh0 + dir * GG;
      } else {
        const int s = (l - 1) * 2 + dir;
        Wih = WihR + s * GG * 20;   Whh = WhhR + s * GG * HH;
        bih = bihR + s * GG;        bhh = bhhR + s * GG;
      }
      pre_gemm_kernel<<<16384, 128, 0, stream>>>(
          (l == 0) ? 0 : 1, K, Kp, xyz, shapes, amino, Xin, Wih, bih, bhh, pre);
      recur_kernel<<<4, 256, 0, stream>>>(pre, Whh, Xout, dir);
    }
  }

  proj_kernel<<<(TT * BB) / 256, 256, 0, stream>>>(xA, projW, projb);

  // fc1: [1024,3072] -> [1024,256], ReLU+BN1
  fc_gemm_kernel<<<32 * 4, 256, 0, stream>>>(
      projb, 3072, 3072, projb, 3072, 3072, fc1_w, 256, y1, 0,
      bn1_g, bn1_b, bn1_m, bn1_v);
  // fc2: concat(y1[1024,256] | seqvec[1024,1024]) -> [1024,1024], ReLU+BN2
  fc_gemm_kernel<<<32 * 16, 256, 0, stream>>>(
      y1, 256, 256, seqvec, 1024, 1280, fc2_w, 1024, lat, 0,
      bn2_g, bn2_b, bn2_m, bn2_v);
  // fc3: [1024,1024] -> [1024,512], ReLU+BN3
  fc_gemm_kernel<<<32 * 8, 256, 0, stream>>>(
      lat, 1024, 1024, lat, 1024, 1024, fc3_w, 512, y3, 0,
      bn3_g, bn3_b, bn3_m, bn3_v);
  // fc4: [1024,512] -> [1024,40] + bias -> d_out
  fc_gemm_kernel<<<32 * 1, 256, 0, stream>>>(
      y3, 512, 512, y3, 512, 512, fc4_w, 40, out, 1,
      fc4_b, fc4_b, fc4_b, fc4_b);
}